// PersonalityMamba_56538949484956
// MI455X (gfx1250) — compile-verified
//
#include <hip/hip_runtime.h>
#include <math.h>

// ---------------------------------------------------------------------------
// PersonalityMamba forward for MI455X (gfx1250, wave32).
// GEMMs use V_WMMA_F32_16X16X4_F32 (native fp32 matrix pipe, exact precision).
// LDS staging: uniform-branch fast path with float4 (global_load_b128 /
// ds_store_b128), scalar guarded path only for edge tiles.
// Selective scan parallelized over B*E*N=16384 recurrences with shfl reduce.
// ---------------------------------------------------------------------------

typedef float v2f __attribute__((ext_vector_type(2)));
typedef float v8f __attribute__((ext_vector_type(8)));

#define KBLK 32

// Generic tiled GEMM: C[M,N] = A[M,K](lda) @ W[K,N](ldb) [+bias] [+res] [act]
// One wave (32 threads) per 16x16 output tile. act: 0=none, 1=softplus.
__global__ __launch_bounds__(32)
void gemm_f32_wmma(const float* __restrict__ A, int lda,
                   const float* __restrict__ W, int ldb,
                   const float* __restrict__ bias,
                   const float* __restrict__ res,
                   float* __restrict__ C, int ldc,
                   int M, int N, int K, int act)
{
    __shared__ float sA[16][KBLK + 4];
    __shared__ float sB[KBLK][16 + 4];

    const int lane = threadIdx.x;      // 0..31
    const int l16  = lane & 15;
    const int half = lane >> 4;        // half-wave: K pair selector
    const int mtile = blockIdx.y * 16;
    const int ntile = blockIdx.x * 16;

    const bool fullM = (mtile + 16 <= M);
    const bool fullN = (ntile + 16 <= N);

    v8f acc = {0.f,0.f,0.f,0.f,0.f,0.f,0.f,0.f};

    for (int k0 = 0; k0 < K; k0 += KBLK) {
        const bool fullK = (k0 + KBLK <= K);

        // ---- Stage A tile 16 x KBLK ----
        if (fullM & fullK) {
            // Vectorized: 128 float4 = 512 floats; 4 float4 per lane.
            const float* Ab = A + (size_t)mtile * lda + k0;
            #pragma unroll
            for (int i = 0; i < 4; ++i) {
                int idx = lane + i * 32;          // 0..127 float4 slots
                int r = idx >> 3;                 // 8 float4 per 32-float row
                int c = (idx & 7) * 4;
                float4 v = *reinterpret_cast<const float4*>(Ab + (size_t)r * lda + c);
                *reinterpret_cast<float4*>(&sA[r][c]) = v;
            }
        } else {
            #pragma unroll
            for (int i = 0; i < 16; ++i) {
                int idx = lane + i * 32;
                int r = idx >> 5, c = idx & 31;
                int gr = mtile + r, gc = k0 + c;
                sA[r][c] = (gr < M && gc < K) ? A[(size_t)gr * lda + gc] : 0.0f;
            }
        }

        // ---- Stage B tile KBLK x 16 ----
        if (fullN & fullK) {
            const float* Wb = W + (size_t)k0 * ldb + ntile;
            #pragma unroll
            for (int i = 0; i < 4; ++i) {
                int idx = lane + i * 32;          // 0..127 float4 slots
                int r = idx >> 2;                 // 4 float4 per 16-float row
                int c = (idx & 3) * 4;
                float4 v = *reinterpret_cast<const float4*>(Wb + (size_t)r * ldb + c);
                *reinterpret_cast<float4*>(&sB[r][c]) = v;
            }
        } else {
            #pragma unroll
            for (int i = 0; i < 16; ++i) {
                int idx = lane + i * 32;
                int r = idx >> 4, c = idx & 15;
                int gr = k0 + r, gc = ntile + c;
                sB[r][c] = (gr < K && gc < N) ? W[(size_t)gr * ldb + gc] : 0.0f;
            }
        }
        __syncthreads();

        // 8 x WMMA f32 16x16x4. Fragment layout per CDNA5 ISA 7.12.2:
        //   A (16x4): lanes0-15 hold M=lane {K=k,k+1}, lanes16-31 {K=k+2,k+3}
        //   B (4x16): lanes0-15 hold N=lane {K=k,k+1}, lanes16-31 {K=k+2,k+3}
        #pragma unroll
        for (int kk = 0; kk < KBLK; kk += 4) {
            v2f a, b;
            a.x = sA[l16][kk + 2 * half + 0];
            a.y = sA[l16][kk + 2 * half + 1];
            b.x = sB[kk + 2 * half + 0][l16];
            b.y = sB[kk + 2 * half + 1][l16];
            acc = __builtin_amdgcn_wmma_f32_16x16x4_f32(
                false, a, false, b, (short)0, acc, false, false);
        }
        __syncthreads();
    }

    // C/D layout: v[i]: lanes0-15 -> M=i, lanes16-31 -> M=i+8; N = l16
    #pragma unroll
    for (int i = 0; i < 8; ++i) {
        int row = mtile + i + half * 8;
        int col = ntile + l16;
        if (row < M && col < N) {
            float v = acc[i];
            if (bias) v += bias[col];
            if (res)  v += res[(size_t)row * ldc + col];
            if (act == 1) v = (v > 20.0f) ? v : log1pf(expf(v));   // softplus
            C[(size_t)row * ldc + col] = v;
        }
    }
}

// LayerNorm over `cols`, one block per row. Population variance, eps=1e-5.
__global__ __launch_bounds__(256)
void ln_kernel(const float* __restrict__ x, const float* __restrict__ g,
               const float* __restrict__ b, float* __restrict__ o, int cols)
{
    __shared__ float s1[8], s2[8], stats[2];
    const int row = blockIdx.x;
    const float* xr = x + (size_t)row * cols;
    float a = 0.f, q = 0.f;
    for (int c = threadIdx.x; c < cols; c += 256) { float v = xr[c]; a += v; q += v * v; }
    #pragma unroll
    for (int off = 16; off > 0; off >>= 1) {
        a += __shfl_xor(a, off, 32);
        q += __shfl_xor(q, off, 32);
    }
    if ((threadIdx.x & 31) == 0) { s1[threadIdx.x >> 5] = a; s2[threadIdx.x >> 5] = q; }
    __syncthreads();
    if (threadIdx.x == 0) {
        float A2 = 0.f, Q2 = 0.f;
        for (int i = 0; i < 8; ++i) { A2 += s1[i]; Q2 += s2[i]; }
        float mean = A2 / cols;
        float var  = Q2 / cols - mean * mean;
        stats[0] = mean;
        stats[1] = 1.0f / sqrtf(var + 1e-5f);
    }
    __syncthreads();
    float mean = stats[0], inv = stats[1];
    float* orow = o + (size_t)row * cols;
    for (int c = threadIdx.x; c < cols; c += 256)
        orow[c] = (xr[c] - mean) * inv * g[c] + b[c];
}

// Causal depthwise conv (K=4) on u = xz[..., :E], + bias, + SiLU.
__global__ __launch_bounds__(256)
void conv_silu_kernel(const float* __restrict__ xz, const float* __restrict__ cw,
                      const float* __restrict__ cb, float* __restrict__ uo,
                      int B, int T, int E)
{
    int idx = blockIdx.x * blockDim.x + threadIdx.x;
    if (idx >= B * T * E) return;
    int e = idx % E;
    int t = (idx / E) % T;
    int b = idx / (E * T);
    const float* base = xz + (size_t)b * T * 2 * E;
    float acc = cb[e];
    #pragma unroll
    for (int k = 0; k < 4; ++k) {
        int tt = t - 3 + k;
        if (tt >= 0) acc += base[(size_t)tt * 2 * E + e] * cw[e * 4 + k];
    }
    uo[idx] = acc / (1.0f + expf(-acc));   // silu
}

// Selective scan. One thread per (b, e, n); 16-lane shfl reduce for sum_n.
__global__ __launch_bounds__(256)
void scan_kernel(const float* __restrict__ dt, const float* __restrict__ u,
                 const float* __restrict__ dbc, const float* __restrict__ A_log,
                 const float* __restrict__ Dp, float* __restrict__ y,
                 int B, int T, int E)
{
    const int gid = blockIdx.x * blockDim.x + threadIdx.x;
    const int n   = gid & 15;
    const int ch  = gid >> 4;          // b*E + e
    if (ch >= B * E) return;
    const int b = ch / E;
    const int e = ch % E;

    const float a  = -expf(A_log[e * 16 + n]);
    const float dp = Dp[e];
    float h = 0.0f;

    const float* dt_p  = dt  + (size_t)b * T * E + e;
    const float* u_p   = u   + (size_t)b * T * E + e;
    const float* dbc_p = dbc + (size_t)b * T * 40;
    float* y_p = y + (size_t)b * T * E + e;

    for (int t = 0; t < T; ++t) {
        if ((t & 15) == 0 && t + 16 < T) {
            __builtin_prefetch(dt_p  + (size_t)(t + 16) * E,  0, 1);
            __builtin_prefetch(dbc_p + (size_t)(t + 16) * 40, 0, 1);
        }
        float dtv = dt_p[(size_t)t * E];
        float uv  = u_p [(size_t)t * E];
        float bc  = dbc_p[(size_t)t * 40 + 8  + n];
        float cc  = dbc_p[(size_t)t * 40 + 24 + n];
        h = expf(dtv * a) * h + dtv * bc * uv;
        float ys = h * cc;
        ys += __shfl_xor(ys, 1, 16);
        ys += __shfl_xor(ys, 2, 16);
        ys += __shfl_xor(ys, 4, 16);
        ys += __shfl_xor(ys, 8, 16);
        if (n == 0) y_p[(size_t)t * E] = ys + dp * uv;
    }
}

// y *= silu(z), z = xz[..., E:2E]
__global__ __launch_bounds__(256)
void gate_kernel(float* __restrict__ y, const float* __restrict__ xz, int total, int E)
{
    int idx = blockIdx.x * blockDim.x + threadIdx.x;
    if (idx >= total) return;
    int e = idx % E;
    int row = idx / E;
    float z = xz[(size_t)row * (2 * E) + E + e];
    y[idx] *= z / (1.0f + expf(-z));
}

__global__ void mean_kernel(const float* __restrict__ per, float* __restrict__ pooled,
                            int B, int T, int D)
{
    int idx = blockIdx.x * blockDim.x + threadIdx.x;
    if (idx >= B * D) return;
    int b = idx / D, d = idx % D;
    float s = 0.f;
    for (int t = 0; t < T; ++t) s += per[((size_t)b * T + t) * D + d];
    pooled[idx] = s / (float)T;
}

__global__ void fc1_kernel(const float* __restrict__ p, const float* __restrict__ w,
                           const float* __restrict__ bias, float* __restrict__ o,
                           int B, int D, int F)
{
    int idx = blockIdx.x * blockDim.x + threadIdx.x;
    if (idx >= B * F) return;
    int b = idx / F, j = idx % F;
    float s = bias[j];
    for (int c = 0; c < D; ++c) s += p[b * D + c] * w[c * F + j];
    o[idx] = s;
}

__global__ void gelu_kernel(float* __restrict__ x, int n)
{
    int i = blockIdx.x * blockDim.x + threadIdx.x;
    if (i < n) {
        float v = x[i];
        x[i] = 0.5f * v * (1.0f + erff(v * 0.7071067811865476f));   // exact gelu
    }
}

__global__ void fc2_kernel(const float* __restrict__ h, const float* __restrict__ w,
                           const float* __restrict__ bias, float* __restrict__ o,
                           int B, int F, int P)
{
    int idx = blockIdx.x * blockDim.x + threadIdx.x;
    if (idx >= B * P) return;
    int b = idx / P, j = idx % P;
    float s = bias[j];
    for (int c = 0; c < F; ++c) s += h[b * F + c] * w[c * P + j];
    o[idx] = 1.0f / (1.0f + expf(-s));
}

extern "C" void kernel_launch(void* const* d_in, const int* in_sizes, int n_in,
                              void* d_out, int out_size, void* d_ws, size_t ws_size,
                              hipStream_t stream)
{
    (void)in_sizes; (void)n_in; (void)out_size; (void)ws_size;
    const float* x        = (const float*)d_in[0];
    const float* proj_w   = (const float*)d_in[1];
    const float* proj_b   = (const float*)d_in[2];
    const float* proj_ln_g= (const float*)d_in[3];
    const float* proj_ln_b= (const float*)d_in[4];
    const float* ln_g     = (const float*)d_in[5];
    const float* ln_b     = (const float*)d_in[6];
    const float* in_w     = (const float*)d_in[7];
    const float* in_b     = (const float*)d_in[8];
    const float* conv_w   = (const float*)d_in[9];
    const float* conv_b   = (const float*)d_in[10];
    const float* xp_w     = (const float*)d_in[11];
    const float* dt_w     = (const float*)d_in[12];
    const float* dt_b     = (const float*)d_in[13];
    const float* A_log    = (const float*)d_in[14];
    const float* Dp       = (const float*)d_in[15];
    const float* out_w    = (const float*)d_in[16];
    const float* out_b    = (const float*)d_in[17];
    const float* fc1_w    = (const float*)d_in[18];
    const float* fc1_b    = (const float*)d_in[19];
    const float* fc_ln_g  = (const float*)d_in[20];
    const float* fc_ln_b  = (const float*)d_in[21];
    const float* fc2_w    = (const float*)d_in[22];
    const float* fc2_b    = (const float*)d_in[23];
    float* out = (float*)d_out;

    const int Bn = 4, T = 2048, Din = 512, D = 128, E = 256, Nst = 16, R = 8,
              L = 2, F = 512, P = 5;
    const int MT = Bn * T;   // 8192 rows

    // Workspace layout (~52 MB of fp32, L2-resident on MI455X)
    float* ws    = (float*)d_ws;
    float* per   = ws;  ws += (size_t)MT * D;
    float* hbuf  = ws;  ws += (size_t)MT * D;
    float* xz    = ws;  ws += (size_t)MT * 2 * E;
    float* uconv = ws;  ws += (size_t)MT * E;
    float* dbc   = ws;  ws += (size_t)MT * 40;
    float* dtb   = ws;  ws += (size_t)MT * E;
    float* ybuf  = ws;  ws += (size_t)MT * E;
    float* pooled= ws;  ws += Bn * D;
    float* h1    = ws;  ws += Bn * F;
    float* h2    = ws;  ws += Bn * F;

    // 1) per = LN(x @ proj_w + proj_b)
    gemm_f32_wmma<<<dim3(D / 16, MT / 16), 32, 0, stream>>>(
        x, Din, proj_w, D, proj_b, nullptr, hbuf, D, MT, D, Din, 0);
    ln_kernel<<<MT, 256, 0, stream>>>(hbuf, proj_ln_g, proj_ln_b, per, D);

    // 2) Mamba blocks
    for (int l = 0; l < L; ++l) {
        const float* lng = ln_g  + (size_t)l * D;
        const float* lnb = ln_b  + (size_t)l * D;
        const float* inw = in_w  + (size_t)l * D * 2 * E;
        const float* inb = in_b  + (size_t)l * 2 * E;
        const float* cw  = conv_w+ (size_t)l * E * 4;
        const float* cb  = conv_b+ (size_t)l * E;
        const float* xpw = xp_w  + (size_t)l * E * 40;
        const float* dtw = dt_w  + (size_t)l * R * E;
        const float* dtbias = dt_b + (size_t)l * E;
        const float* Al  = A_log + (size_t)l * E * Nst;
        const float* Dpl = Dp    + (size_t)l * E;
        const float* ow  = out_w + (size_t)l * E * D;
        const float* ob  = out_b + (size_t)l * D;

        ln_kernel<<<MT, 256, 0, stream>>>(per, lng, lnb, hbuf, D);
        gemm_f32_wmma<<<dim3((2 * E) / 16, MT / 16), 32, 0, stream>>>(
            hbuf, D, inw, 2 * E, inb, nullptr, xz, 2 * E, MT, 2 * E, D, 0);
        conv_silu_kernel<<<(MT * E) / 256, 256, 0, stream>>>(xz, cw, cb, uconv, Bn, T, E);
        gemm_f32_wmma<<<dim3(3, MT / 16), 32, 0, stream>>>(
            uconv, E, xpw, 40, nullptr, nullptr, dbc, 40, MT, 40, E, 0);   // N=40
        gemm_f32_wmma<<<dim3(E / 16, MT / 16), 32, 0, stream>>>(
            dbc, 40, dtw, E, dtbias, nullptr, dtb, E, MT, E, R, 1);        // softplus
        scan_kernel<<<(Bn * E * Nst) / 256, 256, 0, stream>>>(
            dtb, uconv, dbc, Al, Dpl, ybuf, Bn, T, E);
        gate_kernel<<<(MT * E) / 256, 256, 0, stream>>>(ybuf, xz, MT * E, E);
        gemm_f32_wmma<<<dim3(D / 16, MT / 16), 32, 0, stream>>>(
            ybuf, E, ow, D, ob, per, per, D, MT, D, E, 0);                 // + residual
    }

    // 3) Head
    mean_kernel<<<(Bn * D + 255) / 256, 256, 0, stream>>>(per, pooled, Bn, T, D);
    fc1_kernel<<<(Bn * F + 255) / 256, 256, 0, stream>>>(pooled, fc1_w, fc1_b, h1, Bn, D, F);
    ln_kernel<<<Bn, 256, 0, stream>>>(h1, fc_ln_g, fc_ln_b, h2, F);
    gelu_kernel<<<(Bn * F + 255) / 256, 256, 0, stream>>>(h2, Bn * F);
    fc2_kernel<<<1, 32, 0, stream>>>(h2, fc2_w, fc2_b, out, Bn, F, P);
}